// _MemoryLevel_30614526886153
// MI455X (gfx1250) — compile-verified
//
#include <hip/hip_runtime.h>
#include <hip/hip_bf16.h>
#include <math.h>

#define DIM     256
#define NSLOTS  4096
#define HEADS   4
#define RANK    32
#define TOPK    32
#define BATCH   4
#define QKDIM   (HEADS * RANK)   // 128
#define LN_EPS  1e-5f

typedef __attribute__((ext_vector_type(2))) float v2f;
typedef __attribute__((ext_vector_type(8))) float v8f;

static __device__ __forceinline__ float wave_sum(float x) {
#pragma unroll
    for (int off = 16; off > 0; off >>= 1) x += __shfl_xor(x, off, 32);
    return x;
}
static __device__ __forceinline__ float wave_max(float x) {
#pragma unroll
    for (int off = 16; off > 0; off >>= 1) x = fmaxf(x, __shfl_xor(x, off, 32));
    return x;
}

// ---------------------------------------------------------------------------
// Kernel A: val0 = unit_normalize(layer_norm(init_val))     [NSLOTS, DIM]
// One wave32 per row; 8 elements per lane; all reductions via shuffles.
// ---------------------------------------------------------------------------
__global__ void __launch_bounds__(256)
ln_unit_kernel(const float* __restrict__ init_val,
               const float* __restrict__ gamma,
               const float* __restrict__ beta,
               float* __restrict__ val0) {
    const int lane = threadIdx.x & 31;
    const int wave = threadIdx.x >> 5;
    const int row  = blockIdx.x * 8 + wave;
    const float* x = init_val + (size_t)row * DIM;

    float v[8];
    float s = 0.f;
#pragma unroll
    for (int t = 0; t < 8; ++t) { v[t] = x[lane + 32 * t]; s += v[t]; }
    const float mean = wave_sum(s) * (1.f / DIM);

    float s2 = 0.f;
#pragma unroll
    for (int t = 0; t < 8; ++t) { float d = v[t] - mean; s2 += d * d; }
    const float var = wave_sum(s2) * (1.f / DIM);
    const float inv = rsqrtf(var + LN_EPS);

    float nn = 0.f;
#pragma unroll
    for (int t = 0; t < 8; ++t) {
        const int d = lane + 32 * t;
        v[t] = (v[t] - mean) * inv * gamma[d] + beta[d];
        nn += v[t] * v[t];
    }
    const float norm = sqrtf(wave_sum(nn));
    const float r = 1.f / fmaxf(norm, 1e-6f);

    float* o = val0 + (size_t)row * DIM;
#pragma unroll
    for (int t = 0; t < 8; ++t) o[lane + 32 * t] = v[t] * r;
}

// ---------------------------------------------------------------------------
// Kernel B: Qcat = val0 @ Wq_cat, Kcat = val0 @ Wk_cat      [NSLOTS, QKDIM]
// WMMA f32 16x16x4. Block = 16 rows, 4 waves; 16 (mat, col-tile) jobs.
// Head re-indexing folded into the B-fragment gather: col = h*32 + r.
// ---------------------------------------------------------------------------
__global__ void __launch_bounds__(128)
proj_kernel(const float* __restrict__ val0,
            const float* __restrict__ Wq,
            const float* __restrict__ Wk,
            float* __restrict__ Qc,
            float* __restrict__ Kc) {
    const int lane  = threadIdx.x & 31;
    const int wave  = threadIdx.x >> 5;
    const int r0    = blockIdx.x * 16;
    const int mrow  = r0 + (lane & 15);
    const int kbase = (lane < 16) ? 0 : 2;   // A/B 32-bit fragment K split

    for (int job = wave; job < 16; job += 4) {
        const int matId = job >> 3;          // 0 = Q, 1 = K
        const int hc    = job & 7;           // column tile (8 tiles of 16)
        const int col   = hc * 16 + (lane & 15);
        const int h     = col >> 5;
        const int rr    = col & 31;
        const float* W  = matId ? Wk : Wq;
        const float* Wb = W + (size_t)h * DIM * RANK + rr;

        v8f c = {};
#pragma unroll 4
        for (int ks = 0; ks < DIM / 4; ++ks) {
            const int k0 = ks * 4 + kbase;
            v2f a; a.x = val0[(size_t)mrow * DIM + k0];
                   a.y = val0[(size_t)mrow * DIM + k0 + 1];
            v2f b; b.x = Wb[(size_t)k0 * RANK];
                   b.y = Wb[(size_t)(k0 + 1) * RANK];
            c = __builtin_amdgcn_wmma_f32_16x16x4_f32(false, a, false, b,
                                                      (short)0, c, false, false);
        }
        float* O = matId ? Kc : Qc;
        const int rsel = r0 + ((lane >= 16) ? 8 : 0);
#pragma unroll
        for (int v = 0; v < 8; ++v)
            O[(size_t)(rsel + v) * QKDIM + col] = c[v];
    }
}

// ---------------------------------------------------------------------------
// Kernel C (fused): 16-row score stripe in LDS (16*4096*4 = 256KB), then
// per-row top-32 by |score|, signed softmax (one edge per lane, wave32),
// message gather + residual + unit-normalize, broadcast to 4 batches.
// ---------------------------------------------------------------------------
__global__ void __launch_bounds__(128)
scores_topk_kernel(const float* __restrict__ val0,
                   const float* __restrict__ Qc,
                   const float* __restrict__ Kc,
                   float* __restrict__ out) {
    extern __shared__ float sm[];            // [16][NSLOTS] scores
    const int lane  = threadIdx.x & 31;
    const int wave  = threadIdx.x >> 5;
    const int r0    = blockIdx.x * 16;
    const int kbase = (lane < 16) ? 0 : 2;

    // A fragments for this block's 16 Q-rows: 32 k-steps, kept in VGPRs.
    v2f afr[32];
    const int arow = r0 + (lane & 15);
#pragma unroll
    for (int ks = 0; ks < 32; ++ks) {
        afr[ks].x = Qc[(size_t)arow * QKDIM + ks * 4 + kbase];
        afr[ks].y = Qc[(size_t)arow * QKDIM + ks * 4 + kbase + 1];
    }

    // Phase 1: scores stripe, 256 column tiles split over 4 waves.
    for (int ct = wave; ct < NSLOTS / 16; ct += 4) {
        const int brow = ct * 16 + (lane & 15);
        v8f c = {};
#pragma unroll
        for (int ks = 0; ks < 32; ++ks) {
            v2f b; b.x = Kc[(size_t)brow * QKDIM + ks * 4 + kbase];
                   b.y = Kc[(size_t)brow * QKDIM + ks * 4 + kbase + 1];
            c = __builtin_amdgcn_wmma_f32_16x16x4_f32(false, afr[ks], false, b,
                                                      (short)0, c, false, false);
        }
        const int rsel = (lane >= 16) ? 8 : 0;
#pragma unroll
        for (int v = 0; v < 8; ++v)
            sm[(size_t)(rsel + v) * NSLOTS + ct * 16 + (lane & 15)] = c[v];
    }
    __syncthreads();

    // Phase 2: one wave per row (4 rows per wave).
    for (int r = wave; r < 16; r += 4) {
        volatile float* rowp = sm + (size_t)r * NSLOTS;
        const int grow = r0 + r;

        int   sel_idx = 0;
        float sel_s   = 0.f;
        for (int it = 0; it < TOPK; ++it) {
            float best = -1.f; int bidx = 0x7fffffff; float bs = 0.f;
            for (int cidx = lane; cidx < NSLOTS; cidx += 32) {
                const float sv = rowp[cidx];
                const float av = fabsf(sv);
                if (av > best) { best = av; bidx = cidx; bs = sv; }
            }
#pragma unroll
            for (int off = 16; off > 0; off >>= 1) {
                const float ob = __shfl_down(best, off, 32);
                const int   oi = __shfl_down(bidx, off, 32);
                const float os = __shfl_down(bs, off, 32);
                if (ob > best || (ob == best && oi < bidx)) { best = ob; bidx = oi; bs = os; }
            }
            bidx = __shfl(bidx, 0, 32);
            bs   = __shfl(bs, 0, 32);
            if (lane == it) { sel_idx = bidx; sel_s = bs; }
            if (lane == 0) rowp[bidx] = 0.0f;   // claim: |0| won't win again
        }

        // signed softmax over the 32 selected edges (one per lane)
        const float av  = fabsf(sel_s);
        const float m   = wave_max(av);
        const float e   = __expf(av - m);
        const float den = wave_sum(e);
        const float sgn = (sel_s > 0.f) ? 1.f : ((sel_s < 0.f) ? -1.f : 0.f);
        const float w   = sgn * e / den;

        // message + residual
        float acc[8];
        {
            const float* vr = val0 + (size_t)grow * DIM;
#pragma unroll
            for (int t = 0; t < 8; ++t) acc[t] = vr[lane + 32 * t];
        }
        for (int j = 0; j < TOPK; ++j) {
            const float wj = __shfl(w, j, 32);
            const int   ij = __shfl(sel_idx, j, 32);
            const float* vp = val0 + (size_t)ij * DIM;
#pragma unroll
            for (int t = 0; t < 8; ++t) acc[t] += wj * vp[lane + 32 * t];
        }

        // unit normalize
        float nn = 0.f;
#pragma unroll
        for (int t = 0; t < 8; ++t) nn += acc[t] * acc[t];
        const float norm = sqrtf(wave_sum(nn));
        const float ri = 1.f / fmaxf(norm, 1e-6f);
#pragma unroll
        for (int t = 0; t < 8; ++t) acc[t] *= ri;

        // batch-invariant: replicate to all 4 batch slots
#pragma unroll
        for (int b = 0; b < BATCH; ++b) {
            float* o = out + ((size_t)b * NSLOTS + grow) * DIM;
#pragma unroll
            for (int t = 0; t < 8; ++t) o[lane + 32 * t] = acc[t];
        }
    }
}

// ---------------------------------------------------------------------------
extern "C" void kernel_launch(void* const* d_in, const int* in_sizes, int n_in,
                              void* d_out, int out_size, void* d_ws, size_t ws_size,
                              hipStream_t stream) {
    // inputs: 0 batch_size, 1 init_state (unused), 2 init_val,
    //         3 ln_gamma, 4 ln_beta, 5 Wq, 6 Wk
    const float* init_val = (const float*)d_in[2];
    const float* gamma    = (const float*)d_in[3];
    const float* beta     = (const float*)d_in[4];
    const float* Wq       = (const float*)d_in[5];
    const float* Wk       = (const float*)d_in[6];
    float* out = (float*)d_out;

    float* val0 = (float*)d_ws;                    // 4096*256 f32 = 4 MB
    float* Qc   = val0 + (size_t)NSLOTS * DIM;     // 4096*128 f32 = 2 MB
    float* Kc   = Qc   + (size_t)NSLOTS * QKDIM;   // 4096*128 f32 = 2 MB

    ln_unit_kernel<<<NSLOTS / 8, 256, 0, stream>>>(init_val, gamma, beta, val0);
    proj_kernel<<<NSLOTS / 16, 128, 0, stream>>>(val0, Wq, Wk, Qc, Kc);
    scores_topk_kernel<<<NSLOTS / 16, 128, 16 * NSLOTS * sizeof(float), stream>>>(
        val0, Qc, Kc, out);
}